// ChamferLoss_16844861735655
// MI455X (gfx1250) — compile-verified
//
#include <hip/hip_runtime.h>

typedef __attribute__((ext_vector_type(2))) float v2f;
typedef __attribute__((ext_vector_type(8))) float v8f;

#define NBATCH 32
#define NPTS 2048
#define ROWS_PER_BLOCK 128
#define TILE 16
#define CHAM_EPS 1e-16f
#define INF_BITS 0x7F800000u

__global__ __launch_bounds__(256) void chamfer_init_kernel(unsigned* __restrict__ colmin,
                                                           float* __restrict__ out) {
    int i = blockIdx.x * 256 + threadIdx.x;
    if (i < NBATCH * NPTS) colmin[i] = INF_BITS;
    if (i == 0) out[0] = 0.0f;
}

__global__ __launch_bounds__(256) void chamfer_main_kernel(const float* __restrict__ p,
                                                           const float* __restrict__ q,
                                                           unsigned* __restrict__ colmin_g,
                                                           float* __restrict__ out) {
    __shared__ float4   lds_q[NPTS];             // 32 KB: async-staged raw q slab
    __shared__ float4   lds_packed[2][NPTS];     // 64 KB: per-half B-frag + (y2+eps)
    __shared__ unsigned lds_colmin[NPTS];        // 8 KB: per-workgroup column mins
    __shared__ float    lds_x2[ROWS_PER_BLOCK];  // row squared norms
    __shared__ unsigned lds_rowmin[ROWS_PER_BLOCK];

    const int tid  = threadIdx.x;
    const int wave = tid >> 5;
    const int lane = tid & 31;
    const int half = lane >> 4;   // 0: K=0,1 / M=r ; 1: K=2,3 / M=r+8
    const int l15  = lane & 15;

    const int b       = blockIdx.x >> 4;   // 16 row-tiles per batch
    const int rowTile = blockIdx.x & 15;
    const int rowbase = rowTile * ROWS_PER_BLOCK + wave * TILE;

    // ---- Async-stage q[b] (2048 x float4) into LDS: CDNA5 ASYNCcnt path ----
    {
        const float4* qb = (const float4*)q + b * NPTS;
        for (int i = tid; i < NPTS; i += 256) {
            unsigned lds_addr = (unsigned)(unsigned long long)&lds_q[i];
            unsigned long long ga = (unsigned long long)(qb + i);
            asm volatile("global_load_async_to_lds_b128 %0, %1, off"
                         :: "v"(lds_addr), "v"(ga) : "memory");
        }
    }

    for (int j = tid; j < NPTS; j += 256) lds_colmin[j] = INF_BITS;
    if (tid < ROWS_PER_BLOCK) lds_rowmin[tid] = INF_BITS;

    // A fragment: 16 rows x K=4 in fp32, component 0 zeroed (reference drops it).
    // p shape (2,32,2048,4): p_real = p[0] -> offset 0.
    const float4 p4 = ((const float4*)p)[b * NPTS + rowbase + l15];
    v2f afrag;
    afrag.x = half ? p4.z : 0.0f;   // K=0 (zero) / K=2
    afrag.y = half ? p4.w : p4.y;   // K=1       / K=3
    lds_x2[wave * TILE + l15] = p4.y * p4.y + p4.z * p4.z + p4.w * p4.w;

    asm volatile("s_wait_asynccnt 0" ::: "memory");
    __syncthreads();

    // ---- Prepass: pack both lane-half B fragments + (y2 + eps) once per WG ----
    // max(d2,0)+eps == max(d2+eps, eps) exactly, so eps is folded into y2 here.
    for (int i = tid; i < NPTS; i += 256) {
        const float4 q4 = lds_q[i];
        const float y2e = q4.y * q4.y + q4.z * q4.z + q4.w * q4.w + CHAM_EPS;
        lds_packed[0][i] = make_float4(0.0f, q4.y, y2e, 0.0f);  // K=0 (zero), K=1
        lds_packed[1][i] = make_float4(q4.z, q4.w, y2e, 0.0f);  // K=2, K=3
    }
    __syncthreads();

    float x2v[8];
#pragma unroll
    for (int r = 0; r < 8; ++r)
        x2v[r] = lds_x2[wave * TILE + half * 8 + r];

    float rowmin[8];
#pragma unroll
    for (int r = 0; r < 8; ++r) rowmin[r] = __builtin_inff();

    const float4* __restrict__ bbase = &lds_packed[half][0];

    // Sweep all 2048 q points in 16-wide tiles: one f32 WMMA per tile.
#pragma unroll 2
    for (int qt = 0; qt < NPTS / TILE; ++qt) {
        const int col = qt * TILE + l15;
        const float4 bp = bbase[col];           // single ds_load: frag + y2e
        v2f bfrag;
        bfrag.x = bp.x;
        bfrag.y = bp.y;

        v8f c = {};
        // D = A(16x4) * B(4x16) + 0  -> v_wmma_f32_16x16x4_f32
        c = __builtin_amdgcn_wmma_f32_16x16x4_f32(
                /*neg_a=*/false, afrag, /*neg_b=*/false, bfrag,
                /*c_mod=*/(short)0, c, /*reuse_a=*/false, /*reuse_b=*/false);

        float tmin = __builtin_inff();
#pragma unroll
        for (int r = 0; r < 8; ++r) {
            float d2e = x2v[r] + bp.z - 2.0f * c[r];   // d^2 + eps
            float d   = __builtin_amdgcn_sqrtf(__builtin_fmaxf(d2e, CHAM_EPS));
            rowmin[r] = __builtin_fminf(rowmin[r], d);
            tmin      = __builtin_fminf(tmin, d);
        }
        // distances > 0 => uint bit order == float order
        atomicMin(&lds_colmin[col], __float_as_uint(tmin));
    }

#pragma unroll
    for (int r = 0; r < 8; ++r)
        atomicMin(&lds_rowmin[wave * TILE + half * 8 + r], __float_as_uint(rowmin[r]));
    __syncthreads();

    // Merge column mins across row-tiles of this batch.
    for (int j = tid; j < NPTS; j += 256)
        atomicMin(&colmin_g[b * NPTS + j], lds_colmin[j]);

    // Row mins are complete within this workgroup (it swept all q).
    if (tid < ROWS_PER_BLOCK)
        atomicAdd(out, 0.5f * __uint_as_float(lds_rowmin[tid]));
}

__global__ __launch_bounds__(256) void chamfer_colsum_kernel(const unsigned* __restrict__ colmin_g,
                                                             float* __restrict__ out) {
    __shared__ float red[256];
    float s = 0.0f;
    for (int j = threadIdx.x; j < NBATCH * NPTS; j += 256)
        s += 0.5f * __uint_as_float(colmin_g[j]);
    red[threadIdx.x] = s;
    __syncthreads();
    for (int k = 128; k > 0; k >>= 1) {
        if (threadIdx.x < k) red[threadIdx.x] += red[threadIdx.x + k];
        __syncthreads();
    }
    if (threadIdx.x == 0) atomicAdd(out, red[0]);
}

extern "C" void kernel_launch(void* const* d_in, const int* in_sizes, int n_in,
                              void* d_out, int out_size, void* d_ws, size_t ws_size,
                              hipStream_t stream) {
    (void)in_sizes; (void)n_in; (void)out_size; (void)ws_size;
    const float* p = (const float*)d_in[0];   // (2,32,2048,4) f32
    const float* q = (const float*)d_in[1];   // (32,2048,4)  f32
    float* out = (float*)d_out;               // scalar f32
    unsigned* colmin = (unsigned*)d_ws;       // 32*2048 u32 = 256 KB

    const int total = NBATCH * NPTS;          // 65536
    chamfer_init_kernel<<<(total + 255) / 256, 256, 0, stream>>>(colmin, out);
    chamfer_main_kernel<<<NBATCH * (NPTS / ROWS_PER_BLOCK), 256, 0, stream>>>(p, q, colmin, out);
    chamfer_colsum_kernel<<<1, 256, 0, stream>>>(colmin, out);
}